// Laplacian3DLayer_81501299409364
// MI455X (gfx1250) — compile-verified
//
#include <hip/hip_runtime.h>
#include <stdint.h>

// Grid geometry (matches reference)
#define NB 16
#define NX 256
#define NY 128
#define NZ 64
#define TY 16            // y rows per workgroup tile
#define XCHUNK 32        // x planes marched per workgroup
#define PLF (18 * NZ)    // floats per LDS plane: 16 interior + 2 halo rows, 64 z

typedef float f4 __attribute__((ext_vector_type(4)));

// Stencil coefficients, computed in double then cast to f32 (same as reference)
constexpr double DDX = 6.3 / (NX - 1);
constexpr double DDY = 3.1 / (NY - 1);
constexpr double DDZ = 1.5 / (NZ - 1);
constexpr float FCX = (float)(1.0 / (DDX * DDX));
constexpr float FCY = (float)(1.0 / (DDY * DDY));
constexpr float FCZ = (float)(1.0 / (DDZ * DDZ));
constexpr float FCC = (float)(-2.0 * (1.0 / (DDX * DDX) + 1.0 / (DDY * DDY) + 1.0 / (DDZ * DDZ)));

// Flat LDS pointer: low 32 bits are the LDS byte offset (shared aperture is
// {SHARED_BASE[63:32], offset[31:0]}), which is exactly what the async-load
// VDST VGPR wants.
__device__ __forceinline__ uint32_t lds_off32(const void* p) {
  return (uint32_t)(uintptr_t)p;
}

// CDNA5 async copy global -> LDS (16B per lane), tracked by ASYNCcnt.
__device__ __forceinline__ void async_b128(uint32_t lds_off, const float* g) {
  asm volatile("global_load_async_to_lds_b128 %0, %1, off"
               :
               : "v"(lds_off), "v"((uint64_t)(uintptr_t)g)
               : "memory");
}

__device__ __forceinline__ void wait_async0() {
  asm volatile("s_wait_asynccnt 0x0" ::: "memory");
}

// Bring one x-plane (16 interior y rows + 2 halo rows, full z) into LDS slot.
// OOB planes / halo rows are zero-filled with plain LDS stores.
__device__ __forceinline__ void load_plane(int x, int slot, const float* xb,
                                           int y0, int yi, int zi, int tid,
                                           float* lds) {
  float* plane = lds + slot * PLF;
  if (x < 0 || x >= NX) {            // uniform branch: zero plane (x = -1 / NX)
    for (int i = tid; i < PLF; i += 256) plane[i] = 0.0f;
    return;
  }
  const size_t rowbase = (size_t)x * NY;
  // interior: every thread fetches its own 16B
  async_b128(lds_off32(plane + (yi + 1) * NZ + zi * 4),
             xb + (rowbase + (size_t)(y0 + yi)) * NZ + zi * 4);
  // low y halo row
  if (tid < 16) {
    if (y0 > 0) {
      async_b128(lds_off32(plane + tid * 4),
                 xb + (rowbase + (size_t)(y0 - 1)) * NZ + tid * 4);
    } else {
      f4 z4 = {0.0f, 0.0f, 0.0f, 0.0f};
      *(f4*)(plane + tid * 4) = z4;
    }
  } else if (tid < 32) {
    int t = tid - 16;
    if (y0 + TY < NY) {
      async_b128(lds_off32(plane + 17 * NZ + t * 4),
                 xb + (rowbase + (size_t)(y0 + TY)) * NZ + t * 4);
    } else {
      f4 z4 = {0.0f, 0.0f, 0.0f, 0.0f};
      *(f4*)(plane + 17 * NZ + t * 4) = z4;
    }
  }
}

__global__ __launch_bounds__(256) void lap3d_async(const float* __restrict__ xin,
                                                   float* __restrict__ out) {
  __shared__ __align__(16) float lds[4 * PLF];   // 18.4 KB ring of 4 planes

  const int tid = threadIdx.x;
  const int zi = tid & 15;       // 16 lanes cover z = 0..63 as float4
  const int yi = tid >> 4;       // 16 y rows
  const int wg = blockIdx.x;
  const int xc = wg & 7;                 // 8 x chunks
  const int yt = (wg >> 3) & 7;          // 8 y tiles
  const int b  = wg >> 6;                // 16 batches
  const int x0 = xc * XCHUNK;
  const int y0 = yt * TY;

  const float* xb = xin + (size_t)b * NX * NY * NZ;
  float* ob = out + (size_t)b * NX * NY * NZ;

  // prologue: planes x0-1, x0, x0+1 into ring slots 0,1,2
  load_plane(x0 - 1, 0, xb, y0, yi, zi, tid, lds);
  load_plane(x0,     1, xb, y0, yi, zi, tid, lds);
  load_plane(x0 + 1, 2, xb, y0, yi, zi, tid, lds);

  const int r = yi + 1;          // interior row index in plane (halo at 0 / 17)
  const int z = zi * 4;

  for (int ix = 0; ix < XCHUNK; ++ix) {
    const int xx = x0 + ix;

    wait_async0();               // my async copies have landed in LDS
    __syncthreads();             // everyone's copies landed; ring slot is free

    // prefetch plane x+2 while we compute plane x
    load_plane(xx + 2, (ix + 3) & 3, xb, y0, yi, zi, tid, lds);

    const float* pc = lds + ((ix + 1) & 3) * PLF;   // center  (x)
    const float* pm = lds + ((ix    ) & 3) * PLF;   // x - 1
    const float* pp = lds + ((ix + 2) & 3) * PLF;   // x + 1

    f4 c  = *(const f4*)(pc + r * NZ + z);
    f4 xm = *(const f4*)(pm + r * NZ + z);
    f4 xq = *(const f4*)(pp + r * NZ + z);
    f4 ym = *(const f4*)(pc + (r - 1) * NZ + z);
    f4 yq = *(const f4*)(pc + (r + 1) * NZ + z);
    float zl = (zi == 0)  ? 0.0f : pc[r * NZ + z - 1];   // speculative read stays in-bounds
    float zh = (zi == 15) ? 0.0f : pc[r * NZ + z + 4];

    f4 o;
    o.x = FCC * c.x + FCX * (xm.x + xq.x) + FCY * (ym.x + yq.x) + FCZ * (zl  + c.y);
    o.y = FCC * c.y + FCX * (xm.y + xq.y) + FCY * (ym.y + yq.y) + FCZ * (c.x + c.z);
    o.z = FCC * c.z + FCX * (xm.z + xq.z) + FCY * (ym.z + yq.z) + FCZ * (c.y + c.w);
    o.w = FCC * c.w + FCX * (xm.w + xq.w) + FCY * (ym.w + yq.w) + FCZ * (c.z + zh);

    // write-once output: non-temporal so it doesn't evict reusable input from L2
    __builtin_nontemporal_store(o, (f4*)(ob + ((size_t)xx * NY + (size_t)(y0 + yi)) * NZ + z));
  }
}

extern "C" void kernel_launch(void* const* d_in, const int* in_sizes, int n_in,
                              void* d_out, int out_size, void* d_ws, size_t ws_size,
                              hipStream_t stream) {
  const float* xin = (const float*)d_in[0];
  float* out = (float*)d_out;
  (void)in_sizes; (void)n_in; (void)out_size; (void)d_ws; (void)ws_size;

  dim3 grid(NB * (NY / TY) * (NX / XCHUNK));   // 16 * 8 * 8 = 1024 workgroups
  dim3 block(256);                             // 8 wave32 per WG
  lap3d_async<<<grid, block, 0, stream>>>(xin, out);
}